// NetVLAD_45552423141540
// MI455X (gfx1250) — compile-verified
//
#include <hip/hip_runtime.h>
#include <math.h>

#define NROWS   200000
#define DDIM    128
#define KCL     64
#define CHUNK   16
#define NCHUNKS (NROWS / CHUNK)   // 12500

typedef __attribute__((ext_vector_type(2))) float v2f;
typedef __attribute__((ext_vector_type(8))) float v8f;

__device__ __forceinline__ v8f wmma_f32_16x16x4(v2f a, v2f b, v8f c) {
  // 8 args: (neg_a, A, neg_b, B, c_mod, C, reuse_a, reuse_b)
  return __builtin_amdgcn_wmma_f32_16x16x4_f32(false, a, false, b, (short)0, c,
                                               false, false);
}

// ---------------------------------------------------------------- zero ws ---
__global__ void nv_zero(float* __restrict__ ws, int n) {
  for (int i = threadIdx.x + blockIdx.x * blockDim.x; i < n;
       i += blockDim.x * gridDim.x)
    ws[i] = 0.0f;
}

// ------------------------------------------------- pass 1: column sum(x^2) ---
__global__ void __launch_bounds__(256)
nv_colsumsq(const float* __restrict__ x, float* __restrict__ ssq) {
  const int tid  = threadIdx.x;
  const int c4   = tid & 31;   // float4 column group (covers 128 cols)
  const int rgrp = tid >> 5;   // 8 row-groups per block iteration
  const float4* x4 = (const float4*)x;
  float4 acc = make_float4(0.f, 0.f, 0.f, 0.f);
  for (int row = blockIdx.x * 8 + rgrp; row < NROWS; row += gridDim.x * 8) {
    float4 v = x4[(size_t)row * 32 + c4];
    acc.x += v.x * v.x; acc.y += v.y * v.y;
    acc.z += v.z * v.z; acc.w += v.w * v.w;
  }
  atomicAdd(&ssq[c4 * 4 + 0], acc.x);
  atomicAdd(&ssq[c4 * 4 + 1], acc.y);
  atomicAdd(&ssq[c4 * 4 + 2], acc.z);
  atomicAdd(&ssq[c4 * 4 + 3], acc.w);
}

// ---- pass 2: fused logits/softmax/accumulation, WMMA + async LDS copies ----
// inv[d] is folded into the W tile at preload, so the streamed x tiles go
// to LDS raw via GLOBAL_LOAD_ASYNC_TO_LDS_B128; Sraw = sa^T @ x is scaled by
// inv[d] in the finalize kernel.
__global__ void __launch_bounds__(256)
nv_main(const float* __restrict__ x, const float* __restrict__ W,
        const float* __restrict__ bias, const float* __restrict__ ssq,
        float* __restrict__ Sacc, float* __restrict__ aacc) {
  constexpr int XS = 132;  // padded strides to avoid LDS bank conflicts
  constexpr int WSS = 132;
  constexpr int SS = 65;

  __shared__ float Wsh[KCL * WSS];     // 64 x 128, pre-scaled by inv[d]
  __shared__ float xns[CHUNK * XS];    // 16 x 128 raw x tile (async dest)
  __shared__ float sas[CHUNK * SS];    // 16 x 64 logits -> soft assignment
  __shared__ float inv_s[DDIM];

  const int tid  = threadIdx.x;
  const int wave = tid >> 5;
  const int lane = tid & 31;
  const int hf   = lane >> 4;  // half of the wave: selects K-pair / M+8
  const int lm   = lane & 15;  // M or N index within tile

  // inverse column norms, then W pre-scaled by them
  if (tid < DDIM) inv_s[tid] = 1.0f / fmaxf(sqrtf(ssq[tid]), 1e-12f);
  __syncthreads();
  for (int i = tid; i < KCL * DDIM; i += 256)
    Wsh[(i >> 7) * WSS + (i & 127)] = W[i] * inv_s[i & 127];
  __syncthreads();

  // each wave owns 4 of the 32 (16x16) tiles of S(64x128)
  const int m  = wave >> 1;        // cluster tile 0..3
  const int nb = (wave & 1) * 4;   // first of 4 column tiles
  const v8f vzero = {0.f, 0.f, 0.f, 0.f, 0.f, 0.f, 0.f, 0.f};
  v8f accS[4];
  for (int n = 0; n < 4; ++n) accS[n] = vzero;
  float a_local = 0.0f;
  const float bb = bias[16 * (wave & 3) + lm];  // bias for logits tile lanes

  for (int chunk = blockIdx.x; chunk < NCHUNKS; chunk += gridDim.x) {
    // ---- async copy raw 16x128 x tile directly into LDS (B128/lane) ----
    {
      const unsigned long long gbase =
          (unsigned long long)(const void*)x +
          (unsigned long long)chunk * (CHUNK * DDIM * 4ull);
      for (int i = tid; i < CHUNK * DDIM / 4; i += 256) {  // 512 float4
        const int r  = i >> 5;
        const int c4 = i & 31;
        const unsigned lds = (unsigned)(size_t)&xns[r * XS + c4 * 4];
        const unsigned off = (unsigned)i * 16u;
        asm volatile("global_load_async_to_lds_b128 %0, %1, %2"
                     :: "v"(lds), "v"(off), "s"(gbase)
                     : "memory");
      }
      asm volatile("s_wait_asynccnt 0" ::: "memory");
    }
    if (chunk + gridDim.x < NCHUNKS) {
      const float* nx = x + (size_t)(chunk + gridDim.x) * CHUNK * DDIM;
      __builtin_prefetch(nx + tid * 8, 0, 0);  // global_prefetch_b8 into L2
    }
    __syncthreads();

    // ---- logits(16x64) = x(16x128) @ (W*inv)^T : waves 0..3, 1 tile each ---
    if (wave < 4) {
      const int ct = wave;  // clusters [16*ct, 16*ct+15]
      v8f accL = vzero;
      for (int s = 0; s < 32; ++s) {  // K = 128, 4 per step
        const int k0 = 4 * s + 2 * hf;
        v2f av = {xns[lm * XS + k0], xns[lm * XS + k0 + 1]};
        v2f bv = {Wsh[(16 * ct + lm) * WSS + k0],
                  Wsh[(16 * ct + lm) * WSS + k0 + 1]};
        accL = wmma_f32_16x16x4(av, bv, accL);
      }
      for (int v = 0; v < 8; ++v)
        sas[(v + 8 * hf) * SS + 16 * ct + lm] = accL[v] + bb;
    }
    __syncthreads();

    // ---- softmax over clusters, one lane per row ----
    if (tid < CHUNK) {
      float mx = -3.402823466e38f;
      for (int k = 0; k < KCL; ++k) mx = fmaxf(mx, sas[tid * SS + k]);
      float sum = 0.0f;
      for (int k = 0; k < KCL; ++k) {
        float e = __expf(sas[tid * SS + k] - mx);
        sas[tid * SS + k] = e;
        sum += e;
      }
      const float rs = 1.0f / sum;
      for (int k = 0; k < KCL; ++k) sas[tid * SS + k] *= rs;
    }
    __syncthreads();

    // ---- a[k] += sum_rows soft_assign ----
    if (tid < KCL) {
      float t = 0.0f;
      for (int r = 0; r < CHUNK; ++r) t += sas[r * SS + tid];
      a_local += t;
    }

    // ---- Sraw(64x128) += sa^T(64x16) @ x(16x128), all 8 waves via WMMA ----
    for (int s = 0; s < 4; ++s) {  // inner dim = 16 rows, 4 per step
      const int r0 = 4 * s + 2 * hf;
      v2f av = {sas[r0 * SS + 16 * m + lm],
                sas[(r0 + 1) * SS + 16 * m + lm]};
      for (int n = 0; n < 4; ++n) {
        const int cb = 16 * (nb + n) + lm;
        v2f bv = {xns[r0 * XS + cb], xns[(r0 + 1) * XS + cb]};
        accS[n] = wmma_f32_16x16x4(av, bv, accS[n]);
      }
    }
    __syncthreads();
  }

  // ---- flush per-wave register accumulators ----
  for (int n = 0; n < 4; ++n)
    for (int v = 0; v < 8; ++v) {
      const int kk = 16 * m + v + 8 * hf;      // cluster row
      const int dd = 16 * (nb + n) + lm;       // column
      atomicAdd(&Sacc[kk * DDIM + dd], accS[n][v]);
    }
  if (tid < KCL) atomicAdd(&aacc[tid], a_local);
}

// ---- finalize: vlad = inv[d]*Sraw - a*cent, intra-norm over clusters ----
__global__ void nv_finalize(const float* __restrict__ Sacc,
                            const float* __restrict__ aacc,
                            const float* __restrict__ ssq,
                            const float* __restrict__ cent,
                            float* __restrict__ out) {
  const int d = threadIdx.x;
  if (d >= DDIM) return;
  const float invd = 1.0f / fmaxf(sqrtf(ssq[d]), 1e-12f);
  float ss = 0.0f;
  for (int k = 0; k < KCL; ++k) {
    const float v = invd * Sacc[k * DDIM + d] - aacc[k] * cent[k * DDIM + d];
    ss += v * v;
  }
  const float inv = 1.0f / fmaxf(sqrtf(ss), 1e-12f);
  for (int k = 0; k < KCL; ++k) {
    const float v = invd * Sacc[k * DDIM + d] - aacc[k] * cent[k * DDIM + d];
    out[k * DDIM + d] = v * inv;
  }
}

// --------------------------------------------------------------------------
extern "C" void kernel_launch(void* const* d_in, const int* in_sizes, int n_in,
                              void* d_out, int out_size, void* d_ws,
                              size_t ws_size, hipStream_t stream) {
  const float* x    = (const float*)d_in[0];  // (200000, 128)
  const float* W    = (const float*)d_in[1];  // (64, 128)
  const float* b    = (const float*)d_in[2];  // (64,)
  const float* cent = (const float*)d_in[3];  // (64, 128)
  float* out = (float*)d_out;                 // (1, 64, 128)

  float* ws  = (float*)d_ws;
  float* ssq = ws;                 // 128
  float* S   = ws + 128;           // 64*128 = 8192
  float* a   = ws + 128 + 8192;    // 64
  const int ws_floats = 128 + 8192 + 64;

  nv_zero<<<8, 256, 0, stream>>>(ws, ws_floats);
  nv_colsumsq<<<1024, 256, 0, stream>>>(x, ssq);
  nv_main<<<1024, 256, 0, stream>>>(x, W, b, ssq, S, a);
  nv_finalize<<<1, 128, 0, stream>>>(S, a, ssq, cent, out);
}